// GCN_80762565034379
// MI455X (gfx1250) — compile-verified
//
#include <hip/hip_runtime.h>
#include <hip/hip_bf16.h>

typedef __attribute__((ext_vector_type(2))) float v2f;
typedef __attribute__((ext_vector_type(8))) float v8f;

// Native fp32 L2 atomic add, device scope, no-return (STOREcnt-tracked).
__device__ __forceinline__ void atomic_add_f32(float* p, float v) {
    asm volatile("global_atomic_add_f32 %0, %1, off scope:SCOPE_DEV"
                 : : "v"(p), "v"(v) : "memory");
}

// ---------------------------------------------------------------------------
// Degree / normalization kernels
// ---------------------------------------------------------------------------
__global__ void init_deg_kernel(float* __restrict__ deg, int n) {
    int i = blockIdx.x * blockDim.x + threadIdx.x;
    if (i < n) deg[i] = 1.0f;  // self loop contributes 1
}

__global__ void accum_deg_kernel(const long long* __restrict__ dst,
                                 float* __restrict__ deg, long long e_cnt) {
    long long e = (long long)blockIdx.x * blockDim.x + threadIdx.x;
    if (e < e_cnt) atomic_add_f32(&deg[(int)dst[e]], 1.0f);
}

__global__ void dinv_kernel(const float* __restrict__ deg,
                            float* __restrict__ dinv, int n) {
    int i = blockIdx.x * blockDim.x + threadIdx.x;
    if (i < n) {
        float d = deg[i];
        dinv[i] = (d > 0.0f) ? rsqrtf(d) : 0.0f;
    }
}

// ---------------------------------------------------------------------------
// Dense transform: H = A @ W using exact-fp32 WMMA (V_WMMA_F32_16X16X4_F32).
//   A: [N, K] row-major, W: [K, NC] row-major, H: [N, NC] row-major.
//   One block = 16 rows; NC/16 waves; each wave owns a 16-col tile.
//   A tile (contiguous 16*K floats) is staged into LDS with
//   GLOBAL_LOAD_ASYNC_TO_LDS_B128 using a 16B-chunk XOR swizzle
//   (chunk' = chunk ^ row) so WMMA fragment reads are bank-conflict-free.
// ---------------------------------------------------------------------------
template <int K, int NC, bool RELU>
__global__ void gemm_wmma_kernel(const float* __restrict__ A,
                                 const float* __restrict__ W,
                                 float* __restrict__ H, int n_rows) {
    __shared__ float sA[16 * K];
    constexpr int NTHREADS = 32 * (NC / 16);
    constexpr int CPR = K / 4;  // 16-byte chunks per row

    const int r0  = blockIdx.x * 16;
    const int tid = threadIdx.x;
    const unsigned lds_base = (unsigned)(uintptr_t)&sA[0];

    // Async-copy the 16 x K fp32 tile into LDS (swizzled), ASYNCcnt-tracked.
#pragma unroll
    for (int g = tid; g < 16 * CPR; g += NTHREADS) {
        int r = g / CPR;
        int c = g - r * CPR;
        int row = r0 + r;
        if (row > n_rows - 1) row = n_rows - 1;  // clamp (N%16==0 in practice)
        const float* gp = A + (long long)row * K + c * 4;
        unsigned lds_off = lds_base + (unsigned)(r * (K * 4) + ((c ^ r) << 4));
        asm volatile("global_load_async_to_lds_b128 %0, %1, off"
                     : : "v"(lds_off), "v"(gp) : "memory");
    }
    asm volatile("s_wait_asynccnt 0x0" ::: "memory");
    __syncthreads();

    const int wave = tid >> 5;
    const int lane = tid & 31;
    const int m    = lane & 15;        // M for A-frag, N for B-frag
    const int kh   = (lane >> 4) * 2;  // K sub-offset within the 4-chunk
    const int c0   = wave * 16;

    v8f acc = {};
#pragma unroll
    for (int k = 0; k < K; k += 4) {
        // A fragment (16x4): lane m holds A[m][k+kh], A[m][k+kh+1]
        int chunk = (k >> 2) ^ m;  // undo XOR swizzle
        const float* ap = &sA[m * K + chunk * 4 + kh];
        float ax = ap[0];
        float ay = ap[1];
        if (RELU) { ax = fmaxf(ax, 0.0f); ay = fmaxf(ay, 0.0f); }
        v2f a; a.x = ax; a.y = ay;
        // B fragment (4x16): lane n holds W[k+kh][c0+n], W[k+kh+1][c0+n]
        v2f b;
        b.x = W[(k + kh) * NC + c0 + m];
        b.y = W[(k + kh + 1) * NC + c0 + m];
        acc = __builtin_amdgcn_wmma_f32_16x16x4_f32(
            false, a, false, b, (short)0, acc, false, false);
    }

    // D layout: VGPR r -> row (r + 8*(lane>>4)), col = lane&15
    const int rbase = r0 + (lane >> 4) * 8;
#pragma unroll
    for (int r = 0; r < 8; ++r) {
        int row = rbase + r;
        if (row < n_rows)
            H[(long long)row * NC + c0 + m] = acc[r];
    }
}

// ---------------------------------------------------------------------------
// agg[i][c] = bias[c] + dinv[i]^2 * h[i][c]    (self-loop + bias init)
// ---------------------------------------------------------------------------
template <int F>
__global__ void init_agg_kernel(const float* __restrict__ h,
                                const float* __restrict__ bias,
                                const float* __restrict__ dinv,
                                float* __restrict__ out, int n) {
    constexpr int VPR = F / 4;  // float4 chunks per row
    long long idx = (long long)blockIdx.x * blockDim.x + threadIdx.x;
    long long total = (long long)n * VPR;
    if (idx >= total) return;
    int i  = (int)(idx / VPR);
    int c4 = (int)(idx - (long long)i * VPR) * 4;
    float di = dinv[i];
    float w  = di * di;
    const float4 hv = *(const float4*)(h + (long long)i * F + c4);
    float4 o;
    o.x = bias[c4 + 0] + w * hv.x;
    o.y = bias[c4 + 1] + w * hv.y;
    o.z = bias[c4 + 2] + w * hv.z;
    o.w = bias[c4 + 3] + w * hv.w;
    *(float4*)(out + (long long)i * F + c4) = o;
}

// ---------------------------------------------------------------------------
// Edge scatter: out[dst] += dinv[src]*dinv[dst] * h[src]
//   One wave per edge; each lane handles F/32 contiguous floats.
// ---------------------------------------------------------------------------
template <int F>
__global__ void scatter_edges_kernel(const float* __restrict__ h,
                                     const long long* __restrict__ src,
                                     const long long* __restrict__ dst,
                                     const float* __restrict__ dinv,
                                     float* __restrict__ out, long long e_cnt) {
    constexpr int WPB = 8;  // waves per 256-thread block
    long long e = (long long)blockIdx.x * WPB + (threadIdx.x >> 5);
    if (e >= e_cnt) return;
    const int lane = threadIdx.x & 31;
    const int s = (int)src[e];
    const int d = (int)dst[e];
    const float w = dinv[s] * dinv[d];

    constexpr int V = F / 32;
    const float* hp = h + (long long)s * F + lane * V;
    float* op = out + (long long)d * F + lane * V;
    if constexpr (V == 4) {
        float4 hv = *(const float4*)hp;
        atomic_add_f32(op + 0, hv.x * w);
        atomic_add_f32(op + 1, hv.y * w);
        atomic_add_f32(op + 2, hv.z * w);
        atomic_add_f32(op + 3, hv.w * w);
    } else if constexpr (V == 2) {
        float2 hv = *(const float2*)hp;
        atomic_add_f32(op + 0, hv.x * w);
        atomic_add_f32(op + 1, hv.y * w);
    } else {
#pragma unroll
        for (int i = 0; i < V; ++i) atomic_add_f32(op + i, hp[i] * w);
    }
}

// ---------------------------------------------------------------------------
extern "C" void kernel_launch(void* const* d_in, const int* in_sizes, int n_in,
                              void* d_out, int out_size, void* d_ws, size_t ws_size,
                              hipStream_t stream) {
    constexpr int D = 128, Hh = 128, C = 64;

    const float*     x   = (const float*)d_in[0];
    const long long* ei  = (const long long*)d_in[1];
    const float*     W0  = (const float*)d_in[2];
    const float*     b0  = (const float*)d_in[3];
    const float*     W1  = (const float*)d_in[4];
    const float*     b1  = (const float*)d_in[5];
    const float*     W2  = (const float*)d_in[6];
    const float*     b2  = (const float*)d_in[7];
    float*           out = (float*)d_out;

    const int       N = in_sizes[0] / D;
    const long long E = (long long)in_sizes[1] / 2;
    const long long* src = ei;       // edge_index[0]
    const long long* dst = ei + E;   // edge_index[1]

    // Workspace layout: hA[N*128] | agg[N*128] | deg[N] | dinv[N]
    float* hA   = (float*)d_ws;
    float* agg  = hA + (size_t)N * Hh;
    float* deg  = agg + (size_t)N * Hh;
    float* dinv = deg + N;

    const int T = 256;
    dim3 bN((N + T - 1) / T);
    dim3 bE((unsigned)((E + T - 1) / T));
    dim3 bRow((N + 15) / 16);
    dim3 bEdge((unsigned)((E + 7) / 8));

    // --- normalization (computed once per call) ---
    init_deg_kernel<<<bN, T, 0, stream>>>(deg, N);
    accum_deg_kernel<<<bE, T, 0, stream>>>(dst, deg, E);
    dinv_kernel<<<bN, T, 0, stream>>>(deg, dinv, N);

    // --- layer 0: h = x@W0 ; agg = b0 + dinv^2*h + scatter ---
    gemm_wmma_kernel<D, Hh, false><<<bRow, 32 * (Hh / 16), 0, stream>>>(x, W0, hA, N);
    init_agg_kernel<Hh><<<((long long)N * (Hh / 4) + T - 1) / T, T, 0, stream>>>(hA, b0, dinv, agg, N);
    scatter_edges_kernel<Hh><<<bEdge, T, 0, stream>>>(hA, src, dst, dinv, agg, E);

    // --- layer 1: h = relu(agg)@W1 ; agg = b1 + dinv^2*h + scatter ---
    gemm_wmma_kernel<Hh, Hh, true><<<bRow, 32 * (Hh / 16), 0, stream>>>(agg, W1, hA, N);
    init_agg_kernel<Hh><<<((long long)N * (Hh / 4) + T - 1) / T, T, 0, stream>>>(hA, b1, dinv, agg, N);
    scatter_edges_kernel<Hh><<<bEdge, T, 0, stream>>>(hA, src, dst, dinv, agg, E);

    // --- layer 2: h = agg@W2 ; out = b2 + dinv^2*h + scatter ---
    gemm_wmma_kernel<Hh, C, false><<<bRow, 32 * (C / 16), 0, stream>>>(agg, W2, hA, N);
    init_agg_kernel<C><<<((long long)N * (C / 4) + T - 1) / T, T, 0, stream>>>(hA, b2, dinv, out, N);
    scatter_edges_kernel<C><<<bEdge, T, 0, stream>>>(hA, src, dst, dinv, out, E);
}